// Decoder_14903536517755
// MI455X (gfx1250) — compile-verified
//
#include <hip/hip_runtime.h>
#include <hip/hip_bf16.h>
#include <math.h>

// Problem constants (from reference): V=32000, H=512, L=2, B=32, T=64, U=512
#define VV 32000
#define HH 512
#define LL 2
#define BB 32
#define TT 64
#define UU 512
#define VOUT (VV - 1)   // 31999
#define G3H (3 * HH)    // 1536

typedef __attribute__((ext_vector_type(2))) float v2f;
typedef __attribute__((ext_vector_type(8))) float v8f;

// ---------------------------------------------------------------------------
// Init: h[0] = enc_hid[L-1], h[1] = 0, cur_input = target[:,0], loss = 0
// ---------------------------------------------------------------------------
__global__ void __launch_bounds__(256)
init_state(const int* __restrict__ target, const float* __restrict__ enc_hid,
           float* __restrict__ h, int* __restrict__ cur, float* __restrict__ loss) {
  int idx = blockIdx.x * blockDim.x + threadIdx.x;
  if (idx < BB * HH) {
    h[idx] = enc_hid[(LL - 1) * BB * HH + idx];  // layer 0 <- enc last hidden
    h[BB * HH + idx] = 0.0f;                     // layer 1 <- 0
  }
  if (idx < BB) cur[idx] = target[idx * TT + 0]; // tgt[0] = target[:,0]
  if (idx == 0) *loss = 0.0f;
}

// ---------------------------------------------------------------------------
// Attention + embedding -> x = [emb[cur], att]  (one block per batch row)
// ---------------------------------------------------------------------------
__global__ void __launch_bounds__(256)
attn_embed(const float* __restrict__ enc_out, const float* __restrict__ h,
           const float* __restrict__ emb, const int* __restrict__ cur,
           float* __restrict__ x) {
  __shared__ float q[HH];
  __shared__ float p[UU];
  __shared__ float red[256];
  const int b = blockIdx.x, tid = threadIdx.x;
  const float* h1 = h + (size_t)(1 * BB + b) * HH;   // top-layer hidden
  for (int i = tid; i < HH; i += 256) q[i] = h1[i];
  __syncthreads();

  const float* eb = enc_out + (size_t)b * UU * HH;
  for (int u = tid; u < UU; u += 256) {
    const float* row = eb + (size_t)u * HH;
    float s = 0.0f;
    for (int d = 0; d < HH; ++d) s = fmaf(row[d], q[d], s);
    p[u] = s;
  }
  __syncthreads();

  float m = -INFINITY;
  for (int u = tid; u < UU; u += 256) m = fmaxf(m, p[u]);
  red[tid] = m; __syncthreads();
  for (int s = 128; s > 0; s >>= 1) {
    if (tid < s) red[tid] = fmaxf(red[tid], red[tid + s]);
    __syncthreads();
  }
  m = red[0]; __syncthreads();

  float sum = 0.0f;
  for (int u = tid; u < UU; u += 256) { float e = __expf(p[u] - m); p[u] = e; sum += e; }
  red[tid] = sum; __syncthreads();
  for (int s = 128; s > 0; s >>= 1) {
    if (tid < s) red[tid] += red[tid + s];
    __syncthreads();
  }
  const float inv = 1.0f / red[0];
  __syncthreads();

  // att[d] = (sum_u p[u] * enc[b][u][d]) * inv   -> x[b][H + d]
  for (int d = tid; d < HH; d += 256) {
    float a = 0.0f;
    for (int u = 0; u < UU; ++u) a = fmaf(p[u], eb[(size_t)u * HH + d], a);
    x[(size_t)b * (2 * HH) + HH + d] = a * inv;
  }
  // embedding -> x[b][0:H]
  const float* er = emb + (size_t)cur[b] * HH;
  for (int d = tid; d < HH; d += 256) x[(size_t)b * (2 * HH) + d] = er[d];
}

// ---------------------------------------------------------------------------
// Y(32 x N) = X(32 x K) @ W(N x K)^T + bias, via V_WMMA_F32_16X16X4_F32.
// One wave per 16-column tile; each wave computes both 16-row M tiles so the
// B operand (weights) is loaded once per two WMMAs.
// A layout : lane l -> A[l&15][2*(l>>4) .. +1]          (v2f)
// B layout : lane l -> B[2*(l>>4) .. +1][l&15]          (v2f)
// D layout : vgpr r -> row r + 8*(l>>4), col l&15
// ---------------------------------------------------------------------------
__global__ void __launch_bounds__(128)
gemm_xwT_wmma(const float* __restrict__ X, const float* __restrict__ W,
              const float* __restrict__ bias, float* __restrict__ Y,
              long long ldY, int K, int N) {
  const int lane = threadIdx.x & 31;
  const int wave = threadIdx.x >> 5;
  const int ntile = blockIdx.x * 4 + wave;
  const int n0 = ntile * 16;
  if (n0 >= N) return;                       // wave-uniform

  const int col  = n0 + (lane & 15);
  const int colc = (col < N) ? col : (N - 1); // clamp loads on the tail tile
  const int koff = (lane >> 4) * 2;

  const float* xrow0 = X + (size_t)(lane & 15) * K + koff;
  const float* xrow1 = X + (size_t)(16 + (lane & 15)) * K + koff;
  const float* wrow  = W + (size_t)colc * K + koff;

  v8f acc0 = {};
  v8f acc1 = {};
#pragma unroll 4
  for (int k = 0; k < K; k += 4) {
    v2f a0 = *(const v2f*)(xrow0 + k);
    v2f a1 = *(const v2f*)(xrow1 + k);
    v2f bb = *(const v2f*)(wrow + k);
    acc0 = __builtin_amdgcn_wmma_f32_16x16x4_f32(
        false, a0, false, bb, (short)0, acc0, false, false);
    acc1 = __builtin_amdgcn_wmma_f32_16x16x4_f32(
        false, a1, false, bb, (short)0, acc1, false, false);
  }

  const float bn = bias[colc];
  if (col < N) {
    const int rbase = (lane >> 4) * 8;
#pragma unroll
    for (int r = 0; r < 8; ++r) {
      Y[(size_t)(rbase + r) * ldY + col]      = acc0[r] + bn;  // rows 0..15
      Y[(size_t)(16 + rbase + r) * ldY + col] = acc1[r] + bn;  // rows 16..31
    }
  }
}

// ---------------------------------------------------------------------------
// GRU gate nonlinearity + state update (PyTorch order r,z,n); h in place.
// ---------------------------------------------------------------------------
__global__ void __launch_bounds__(256)
gru_update(const float* __restrict__ gi, const float* __restrict__ gh,
           float* __restrict__ h) {
  int idx = blockIdx.x * blockDim.x + threadIdx.x;
  if (idx >= BB * HH) return;
  int b = idx / HH, d = idx % HH;
  size_t o = (size_t)b * G3H + d;
  float ir = gi[o], iz = gi[o + HH], in_ = gi[o + 2 * HH];
  float hr = gh[o], hz = gh[o + HH], hn  = gh[o + 2 * HH];
  float r = 1.0f / (1.0f + __expf(-(ir + hr)));
  float z = 1.0f / (1.0f + __expf(-(iz + hz)));
  float n = tanhf(in_ + r * hn);
  float hp = h[idx];
  h[idx] = (1.0f - z) * n + z * hp;
}

// ---------------------------------------------------------------------------
// Per-row log-softmax loss at target token + greedy argmax feedback.
// One block per batch row; logits already in d_out at (b, t-1, :).
// ---------------------------------------------------------------------------
__global__ void __launch_bounds__(256)
softmax_loss(const float* __restrict__ out, const int* __restrict__ target,
             int t, int* __restrict__ cur, float* __restrict__ loss) {
  __shared__ float rmax[256];
  __shared__ int   rarg[256];
  __shared__ float rsum[256];
  const int b = blockIdx.x, tid = threadIdx.x;
  const float* row = out + ((size_t)b * (TT - 1) + (t - 1)) * VOUT;

  float m = -INFINITY; int arg = 0;
  for (int n = tid; n < VOUT; n += 256) {
    float v = row[n];
    if (v > m) { m = v; arg = n; }
  }
  rmax[tid] = m; rarg[tid] = arg; __syncthreads();
  for (int s = 128; s > 0; s >>= 1) {
    if (tid < s && rmax[tid + s] > rmax[tid]) {
      rmax[tid] = rmax[tid + s]; rarg[tid] = rarg[tid + s];
    }
    __syncthreads();
  }
  m = rmax[0]; __syncthreads();

  float sum = 0.0f;
  for (int n = tid; n < VOUT; n += 256) sum += __expf(row[n] - m);
  rsum[tid] = sum; __syncthreads();
  for (int s = 128; s > 0; s >>= 1) {
    if (tid < s) rsum[tid] += rsum[tid + s];
    __syncthreads();
  }
  if (tid == 0) {
    int tgt = target[b * TT + t];                     // tgt[t][b]
    float lp = row[tgt] - m - __logf(rsum[0]);
    atomicAdd(loss, -lp * (1.0f / (float)BB));        // mean over batch
    cur[b] = rarg[0];                                 // greedy next input
  }
}

// ---------------------------------------------------------------------------
extern "C" void kernel_launch(void* const* d_in, const int* in_sizes, int n_in,
                              void* d_out, int out_size, void* d_ws, size_t ws_size,
                              hipStream_t stream) {
  const int*   target  = (const int*)  d_in[0];
  const float* enc_out = (const float*)d_in[1];
  const float* enc_hid = (const float*)d_in[2];
  const float* emb     = (const float*)d_in[3];
  const float* w_ih0   = (const float*)d_in[4];
  const float* w_hh0   = (const float*)d_in[5];
  const float* b_ih0   = (const float*)d_in[6];
  const float* b_hh0   = (const float*)d_in[7];
  const float* w_ih1   = (const float*)d_in[8];
  const float* w_hh1   = (const float*)d_in[9];
  const float* b_ih1   = (const float*)d_in[10];
  const float* b_hh1   = (const float*)d_in[11];
  const float* out_w   = (const float*)d_in[12];
  const float* out_b   = (const float*)d_in[13];

  float* out  = (float*)d_out;                               // (B, T-1, V-1)
  float* loss = out + (size_t)BB * (TT - 1) * VOUT;          // trailing scalar

  // Workspace layout (floats)
  float* h  = (float*)d_ws;           // [2][B][H]   = 32768
  float* x  = h  + 2 * BB * HH;       // [B][2H]     = 32768
  float* gi = x  + BB * 2 * HH;       // [B][3H]     = 49152
  float* gh = gi + BB * G3H;          // [B][3H]     = 49152
  int*  cur = (int*)(gh + BB * G3H);  // [B]

  init_state<<<64, 256, 0, stream>>>(target, enc_hid, h, cur, loss);

  const int gateBlocks  = (G3H / 16) / 4;                       // 24
  const int logitTiles  = (VOUT + 15) / 16;                     // 2000
  const int logitBlocks = (logitTiles + 3) / 4;                 // 500

  for (int t = 1; t < TT; ++t) {
    attn_embed<<<BB, 256, 0, stream>>>(enc_out, h, emb, cur, x);

    // GRU layer 0: gi = x @ w_ih0^T + b_ih0 (K=2H); gh = h0 @ w_hh0^T + b_hh0
    gemm_xwT_wmma<<<gateBlocks, 128, 0, stream>>>(x, w_ih0, b_ih0, gi,
                                                  (long long)G3H, 2 * HH, G3H);
    gemm_xwT_wmma<<<gateBlocks, 128, 0, stream>>>(h, w_hh0, b_hh0, gh,
                                                  (long long)G3H, HH, G3H);
    gru_update<<<64, 256, 0, stream>>>(gi, gh, h);

    // GRU layer 1: input is updated h0
    gemm_xwT_wmma<<<gateBlocks, 128, 0, stream>>>(h, w_ih1, b_ih1, gi,
                                                  (long long)G3H, HH, G3H);
    gemm_xwT_wmma<<<gateBlocks, 128, 0, stream>>>(h + BB * HH, w_hh1, b_hh1, gh,
                                                  (long long)G3H, HH, G3H);
    gru_update<<<64, 256, 0, stream>>>(gi, gh, h + BB * HH);

    // Logits: h1 @ out_w^T + out_b -> d_out[b][t-1][:]
    float* ybase = out + (size_t)(t - 1) * VOUT;
    gemm_xwT_wmma<<<logitBlocks, 128, 0, stream>>>(
        h + BB * HH, out_w, out_b, ybase,
        (long long)(TT - 1) * VOUT, HH, VOUT);

    // Loss + greedy feedback
    softmax_loss<<<BB, 256, 0, stream>>>(out, target, t, cur, loss);
  }
}